// OTAlignmentLoss_9448928051790
// MI455X (gfx1250) — compile-verified
//
#include <hip/hip_runtime.h>
#include <cmath>
#include <cstdint>

// ---------------------------------------------------------------------------
// Problem constants (match reference)
// ---------------------------------------------------------------------------
#define BB   512
#define KK   160
#define HH   256
#define KK2  (KK * KK)          // 25600
#define BK   (BB * KK)          // 81920
#define ALPHA_C     0.5f
#define TEMP_C      2.0f
#define MAX_SPAN_C  30
#define L_FGW_C     0.1f
#define L_SPAN_C    0.5f
#define L_CONS_C    0.3f

// workspace offsets (floats)
#define OFF_EN_SL  0
#define OFF_EN_EL  81920
#define OFF_VI_SL  163840
#define OFF_VI_EL  245760
#define OFF_FGW    327680
#define OFF_QA     328192
#define OFF_SPAN   328704
#define OFF_ANS    329216
#define OFF_KL     329728

typedef float v2f __attribute__((ext_vector_type(2)));
typedef float v8f __attribute__((ext_vector_type(8)));
typedef unsigned int v4u __attribute__((ext_vector_type(4)));
typedef int v4i __attribute__((ext_vector_type(4)));
typedef int v8i __attribute__((ext_vector_type(8)));

// ---------------------------------------------------------------------------
// TDM: issue a tensor_load_to_lds of `nelem` fp32 elements (1 row tile).
// D# bitfields per CDNA5 ISA:
//  group0: [1:0]=count=1, [63:32]=lds_addr, [120:64]=global_addr, [127:126]=2
//  group1: [17:16]=data_size(2 -> 4B), dim0=nelem, dim1=1, tile0=nelem,
//          tile1=1, dim0_stride=nelem; groups 2/3 zero (<=2D, no iterate).
// 6-arg builtin form (clang-23 / therock-10.0 headers).
// ---------------------------------------------------------------------------
__device__ __forceinline__ void tdm_load_f32_row(unsigned lds_off,
                                                 const void* gptr,
                                                 unsigned nelem) {
  const unsigned long long ga = (unsigned long long)(uintptr_t)gptr;
  v4u g0;
  g0.x = 1u;                                             // count=1 (valid D#)
  g0.y = lds_off;                                        // LDS byte address
  g0.z = (unsigned)(ga & 0xFFFFFFFFull);                 // global_addr[31:0]
  g0.w = (unsigned)((ga >> 32) & 0x01FFFFFFull)          // global_addr[56:32]
       | (2u << 30);                                     // type=2 ("image")
  v8i g1;
  g1[0] = (int)0x00020000u;                  // wg_mask=0, data_size=2 (4B)
  g1[1] = 0;                                 // no atomic barrier
  g1[2] = (int)((nelem & 0xFFFFu) << 16);    // tensor_dim0 low16  @bits95:80
  g1[3] = (int)((nelem >> 16) | (1u << 16)); // dim0 hi16 | tensor_dim1=1 lo16
  g1[4] = (int)((nelem & 0xFFFFu) << 16);    // dim1 hi16=0 | tile_dim0
  g1[5] = 1;                                 // tile_dim1=1, tile_dim2=0
  g1[6] = (int)nelem;                        // tensor_dim0_stride low32
  g1[7] = 0;                                 // stride hi16 | dim1_stride=0
  v4i g2 = {0, 0, 0, 0};
  v4i g3 = {0, 0, 0, 0};
  v8i g4 = {0, 0, 0, 0, 0, 0, 0, 0};
  __builtin_amdgcn_tensor_load_to_lds(g0, g1, g2, g3, g4, 0);
}

// ---------------------------------------------------------------------------
// Kernel A: QA-head logits. One wave (32 lanes) per row; both heads per row.
// rows 0..BK-1 -> en embeddings, rows BK..2BK-1 -> vi embeddings.
// Memory bound: embeddings read exactly once via b128 loads.
// ---------------------------------------------------------------------------
__global__ __launch_bounds__(256) void qa_logits_kernel(
    const float* __restrict__ en, const float* __restrict__ vi,
    const float* __restrict__ w_s, const float* __restrict__ b_s,
    const float* __restrict__ w_e, const float* __restrict__ b_e,
    float* __restrict__ ws) {
  __shared__ __align__(16) float sWs[HH];
  __shared__ __align__(16) float sWe[HH];
  const int tid  = threadIdx.x;
  sWs[tid] = w_s[tid];
  sWe[tid] = w_e[tid];
  __syncthreads();

  const int wv   = tid >> 5;
  const int lane = tid & 31;
  const long long r = (long long)blockIdx.x * 8 + wv;   // [0, 2*BK)

  const float* emb = (r < BK) ? (en + (size_t)r * HH)
                              : (vi + (size_t)(r - BK) * HH);
  const int h0 = lane * 8;
  const float4 p0 = *(const float4*)(emb + h0);
  const float4 p1 = *(const float4*)(emb + h0 + 4);
  float ds = p0.x * sWs[h0]     + p0.y * sWs[h0 + 1]
           + p0.z * sWs[h0 + 2] + p0.w * sWs[h0 + 3]
           + p1.x * sWs[h0 + 4] + p1.y * sWs[h0 + 5]
           + p1.z * sWs[h0 + 6] + p1.w * sWs[h0 + 7];
  float de = p0.x * sWe[h0]     + p0.y * sWe[h0 + 1]
           + p0.z * sWe[h0 + 2] + p0.w * sWe[h0 + 3]
           + p1.x * sWe[h0 + 4] + p1.y * sWe[h0 + 5]
           + p1.z * sWe[h0 + 6] + p1.w * sWe[h0 + 7];
  #pragma unroll
  for (int off = 16; off > 0; off >>= 1) {
    ds += __shfl_xor(ds, off, 32);
    de += __shfl_xor(de, off, 32);
  }
  if (lane == 0) {
    if (r < BK) {
      ws[OFF_EN_SL + r] = ds + b_s[0];
      ws[OFF_EN_EL + r] = de + b_e[0];
    } else {
      const long long rr = r - BK;
      ws[OFF_VI_SL + rr] = ds + b_s[0];
      ws[OFF_VI_EL + rr] = de + b_e[0];
    }
  }
}

// ---------------------------------------------------------------------------
// Block-wide sum reduction helper (256 threads).
// ---------------------------------------------------------------------------
__device__ __forceinline__ float blockReduceSum256(float v, float* s) {
  const int tid = threadIdx.x;
  __syncthreads();
  s[tid] = v;
  __syncthreads();
  #pragma unroll
  for (int off = 128; off > 0; off >>= 1) {
    if (tid < off) s[tid] += s[tid + off];
    __syncthreads();
  }
  return s[0];
}

// ---------------------------------------------------------------------------
// Kernel B: FGW loss, one workgroup per batch element.
// gamma / D_en / D_vi staged into LDS (300 KB of the 320 KB WGP pool) by the
// Tensor Data Mover (3 tensor_load_to_lds from wave 0, s_wait_tensorcnt).
// gw3 = <D_en @ gamma, gamma @ D_vi> with V_WMMA_F32_16X16X4_F32:
// 25 blocks of 2x2 tiles (32x32) over 8 waves; each k-step feeds 8 WMMAs
// from 4 A-fragments + 8 B-words (2x operand reuse vs per-tile scheme).
// ---------------------------------------------------------------------------
__global__ __launch_bounds__(256) void fgw_kernel(
    const float* __restrict__ gamma, const float* __restrict__ Den,
    const float* __restrict__ Dvi, const float* __restrict__ Mg,
    float* __restrict__ ws) {
  __shared__ __align__(16) float sG[KK2];   // gamma
  __shared__ __align__(16) float sA[KK2];   // D_en
  __shared__ __align__(16) float sBm[KK2];  // D_vi
  __shared__ float sP[KK];
  __shared__ float sQ[KK];
  __shared__ float sRed[256];

  const int b   = blockIdx.x;
  const int tid = threadIdx.x;
  const size_t base = (size_t)b * KK2;

  // ---- TDM staging: wave 0 issues 3 DMA descriptors, waits TENSORcnt ----
  if (tid < 32) {
    tdm_load_f32_row((unsigned)(uintptr_t)sG,  gamma + base, KK2);
    tdm_load_f32_row((unsigned)(uintptr_t)sA,  Den + base,   KK2);
    tdm_load_f32_row((unsigned)(uintptr_t)sBm, Dvi + base,   KK2);
    __builtin_amdgcn_s_wait_tensorcnt(0);
  }
  __syncthreads();

  // marginals p_i (row sums), q_j (col sums) of gamma
  if (tid < KK) {
    float ps = 0.f, qs = 0.f;
    for (int j = 0; j < KK; ++j) {
      ps += sG[tid * KK + j];
      qs += sG[j * KK + tid];
    }
    sP[tid] = ps;
    sQ[tid] = qs;
  }

  // w = sum(M .* gamma): stream M from global once
  float wpart = 0.f;
  for (int idx = tid; idx < KK2; idx += 256) wpart += Mg[base + idx] * sG[idx];
  __syncthreads();

  // gw1 = sum D_en^2[i,j] p_i p_j ; gw2 = sum D_vi^2[i,j] q_i q_j
  float g1 = 0.f, g2 = 0.f;
  for (int idx = tid; idx < KK2; idx += 256) {
    const int i = idx / KK, j = idx % KK;
    const float a = sA[idx];
    const float c = sBm[idx];
    g1 += a * a * sP[i] * sP[j];
    g2 += c * c * sQ[i] * sQ[j];
  }

  // gw3 via dual GEMM in WMMA fp32
  const int lane = tid & 31;
  const int wv   = tid >> 5;
  const int half = lane >> 4;   // 0: K pair {0,1}, 1: K pair {2,3}
  const int l16  = lane & 15;

  float gw3_local = 0.f;
  for (int blk = wv; blk < 25; blk += 8) {
    const int i0 = (blk / 5) * 32;
    const int j0 = (blk % 5) * 32;
    v8f tT00 = {}, tT01 = {}, tT10 = {}, tT11 = {};
    v8f tU00 = {}, tU01 = {}, tU10 = {}, tU11 = {};
    #pragma unroll 2
    for (int k0 = 0; k0 < KK; k0 += 4) {
      const int ka = k0 + 2 * half;
      // A fragments (two row-tiles each for T and U): contiguous -> b64
      const float2 aT0v = *(const float2*)(&sA[(i0 + l16) * KK + ka]);
      const float2 aT1v = *(const float2*)(&sA[(i0 + 16 + l16) * KK + ka]);
      const float2 aU0v = *(const float2*)(&sG[(i0 + l16) * KK + ka]);
      const float2 aU1v = *(const float2*)(&sG[(i0 + 16 + l16) * KK + ka]);
      v2f aT0; aT0.x = aT0v.x; aT0.y = aT0v.y;
      v2f aT1; aT1.x = aT1v.x; aT1.y = aT1v.y;
      v2f aU0; aU0.x = aU0v.x; aU0.y = aU0v.y;
      v2f aU1; aU1.x = aU1v.x; aU1.y = aU1v.y;
      // B fragments (two col-tiles each), 4x16 N-striped over lanes
      v2f bT0; bT0.x = sG [(ka)     * KK + j0 + l16];
               bT0.y = sG [(ka + 1) * KK + j0 + l16];
      v2f bT1; bT1.x = sG [(ka)     * KK + j0 + 16 + l16];
               bT1.y = sG [(ka + 1) * KK + j0 + 16 + l16];
      v2f bU0; bU0.x = sBm[(ka)     * KK + j0 + l16];
               bU0.y = sBm[(ka + 1) * KK + j0 + l16];
      v2f bU1; bU1.x = sBm[(ka)     * KK + j0 + 16 + l16];
               bU1.y = sBm[(ka + 1) * KK + j0 + 16 + l16];
      tT00 = __builtin_amdgcn_wmma_f32_16x16x4_f32(false, aT0, false, bT0,
                                                   (short)0, tT00, false, false);
      tT01 = __builtin_amdgcn_wmma_f32_16x16x4_f32(false, aT0, false, bT1,
                                                   (short)0, tT01, false, false);
      tT10 = __builtin_amdgcn_wmma_f32_16x16x4_f32(false, aT1, false, bT0,
                                                   (short)0, tT10, false, false);
      tT11 = __builtin_amdgcn_wmma_f32_16x16x4_f32(false, aT1, false, bT1,
                                                   (short)0, tT11, false, false);
      tU00 = __builtin_amdgcn_wmma_f32_16x16x4_f32(false, aU0, false, bU0,
                                                   (short)0, tU00, false, false);
      tU01 = __builtin_amdgcn_wmma_f32_16x16x4_f32(false, aU0, false, bU1,
                                                   (short)0, tU01, false, false);
      tU10 = __builtin_amdgcn_wmma_f32_16x16x4_f32(false, aU1, false, bU0,
                                                   (short)0, tU10, false, false);
      tU11 = __builtin_amdgcn_wmma_f32_16x16x4_f32(false, aU1, false, bU1,
                                                   (short)0, tU11, false, false);
    }
    #pragma unroll
    for (int v = 0; v < 8; ++v) {
      gw3_local += tT00[v] * tU00[v] + tT01[v] * tU01[v]
                 + tT10[v] * tU10[v] + tT11[v] * tU11[v];
    }
  }

  const float w   = blockReduceSum256(wpart, sRed);
  const float gw1 = blockReduceSum256(g1, sRed);
  const float gw2 = blockReduceSum256(g2, sRed);
  const float gw3 = blockReduceSum256(gw3_local, sRed);
  if (tid == 0) {
    ws[OFF_FGW + b] = ALPHA_C * (gw1 + gw2 - 2.f * gw3) + (1.f - ALPHA_C) * w;
  }
}

// ---------------------------------------------------------------------------
// Kernel C: span decode + CE + KL, one workgroup per batch element.
// ---------------------------------------------------------------------------
__global__ __launch_bounds__(256) void span_kernel(
    const float* __restrict__ gamma, const int* __restrict__ en_start,
    const int* __restrict__ en_end, float* __restrict__ ws) {
  __shared__ float sVS[KK];
  __shared__ float sRF[256];
  __shared__ int   sRI[256];

  const int b   = blockIdx.x;
  const int tid = threadIdx.x;
  const int s_raw = en_start[b];
  const int e_raw = en_end[b];
  const int s_lab = min(max(s_raw, 0), KK - 1);
  const int e_lab = min(max(e_raw, 0), KK - 1);
  const int s_dec = s_lab;
  const int e_dec = max(s_dec, e_lab);
  const size_t gbase = (size_t)b * KK2;

  // vi_score[j] = sum_{i in [s_dec, e_dec]} gamma[b,i,j]
  if (tid < KK) {
    float acc = 0.f;
    for (int i = s_dec; i <= e_dec; ++i)
      acc += gamma[gbase + (size_t)i * KK + tid];
    sVS[tid] = acc;
  }
  __syncthreads();

  // banded argmax over (i,j), 0 <= j-i <= MAX_SPAN, row-major first-max ties
  float bestv = -INFINITY;
  int   bestf = 0x7fffffff;
  if (tid < KK) {
    const int i = tid;
    const int jmax = min(i + MAX_SPAN_C, KK - 1);
    for (int j = i; j <= jmax; ++j) {
      const float v = sVS[i] + sVS[j];
      if (v > bestv) { bestv = v; bestf = i * KK + j; }
    }
  }
  sRF[tid] = bestv;
  sRI[tid] = bestf;
  __syncthreads();
  for (int off = 128; off > 0; off >>= 1) {
    if (tid < off) {
      const float v2 = sRF[tid + off];
      const int   f2 = sRI[tid + off];
      if (v2 > sRF[tid] || (v2 == sRF[tid] && f2 < sRI[tid])) {
        sRF[tid] = v2;
        sRI[tid] = f2;
      }
    }
    __syncthreads();
  }
  int ps = sRI[0] / KK;
  int pe = sRI[0] % KK;
  if (s_raw == 0 && e_raw == 0) { ps = 0; pe = 0; }

  // logZ(row * invT) via block reductions (uniform control flow)
  auto logZ = [&](const float* __restrict__ row, float invT) -> float {
    const float x = (tid < KK) ? row[tid] * invT : -INFINITY;
    __syncthreads();
    sRF[tid] = x;
    __syncthreads();
    for (int off = 128; off > 0; off >>= 1) {
      if (tid < off) sRF[tid] = fmaxf(sRF[tid], sRF[tid + off]);
      __syncthreads();
    }
    const float m = sRF[0];
    __syncthreads();
    sRF[tid] = (tid < KK) ? __expf(x - m) : 0.f;
    __syncthreads();
    for (int off = 128; off > 0; off >>= 1) {
      if (tid < off) sRF[tid] += sRF[tid + off];
      __syncthreads();
    }
    const float z = m + __logf(sRF[0]);
    __syncthreads();
    return z;
  };

  const float* ensl = ws + OFF_EN_SL + (size_t)b * KK;
  const float* enel = ws + OFF_EN_EL + (size_t)b * KK;
  const float* visl = ws + OFF_VI_SL + (size_t)b * KK;
  const float* viel = ws + OFF_VI_EL + (size_t)b * KK;

  const float z_en_s = logZ(ensl, 1.f);
  const float z_en_e = logZ(enel, 1.f);
  const float z_vi_s = logZ(visl, 1.f);
  const float z_vi_e = logZ(viel, 1.f);
  const float invT = 1.f / TEMP_C;
  const float zT_en_s = logZ(ensl, invT);
  const float zT_en_e = logZ(enel, invT);
  const float zT_vi_s = logZ(visl, invT);
  const float zT_vi_e = logZ(viel, invT);

  // KL terms: sum_k exp(en_lp)*(en_lp - vi_lp), start + end heads
  float klp = 0.f;
  if (tid < KK) {
    const float es = ensl[tid] * invT - zT_en_s;
    const float vs = visl[tid] * invT - zT_vi_s;
    klp += __expf(es) * (es - vs);
    const float ee2 = enel[tid] * invT - zT_en_e;
    const float ve2 = viel[tid] * invT - zT_vi_e;
    klp += __expf(ee2) * (ee2 - ve2);
  }
  const float klb = blockReduceSum256(klp, sRF);

  if (tid == 0) {
    const float ce_en_s = -(ensl[s_lab] - z_en_s);
    const float ce_en_e = -(enel[e_lab] - z_en_e);
    const float ce_vi_s = -(visl[ps] - z_vi_s);
    const float ce_vi_e = -(viel[pe] - z_vi_e);
    const float ansf = (s_raw > 0 || e_raw > 0) ? 1.f : 0.f;
    ws[OFF_QA   + b] = 0.5f * (ce_en_s + ce_en_e);
    ws[OFF_SPAN + b] = ansf * 0.5f * (ce_vi_s + ce_vi_e);
    ws[OFF_ANS  + b] = ansf;
    ws[OFF_KL   + b] = klb;
  }
}

// ---------------------------------------------------------------------------
// Kernel D: final reduction over B and scalar combination.
// ---------------------------------------------------------------------------
__global__ __launch_bounds__(256) void finalize_kernel(
    const float* __restrict__ ws, float* __restrict__ out) {
  __shared__ float sRed[256];
  const int tid = threadIdx.x;
  float qa = 0.f, fgw = 0.f, span = 0.f, ans = 0.f, kl = 0.f;
  for (int b = tid; b < BB; b += 256) {
    qa   += ws[OFF_QA   + b];
    fgw  += ws[OFF_FGW  + b];
    span += ws[OFF_SPAN + b];
    ans  += ws[OFF_ANS  + b];
    kl   += ws[OFF_KL   + b];
  }
  qa   = blockReduceSum256(qa, sRed);
  fgw  = blockReduceSum256(fgw, sRed);
  span = blockReduceSum256(span, sRed);
  ans  = blockReduceSum256(ans, sRed);
  kl   = blockReduceSum256(kl, sRed);
  if (tid == 0) {
    const float l_qa   = qa / (float)BB;
    const float l_fgw  = fgw / (float)BB;
    const float l_span = (ans > 0.f) ? span / fmaxf(ans, 1.f) : 0.f;
    const float l_cons = (TEMP_C * TEMP_C) * kl / (2.f * (float)BB);
    const float total  = l_qa + L_FGW_C * l_fgw + L_SPAN_C * l_span
                       + L_CONS_C * l_cons;
    out[0] = total;
    out[1] = l_qa;
    out[2] = l_fgw;
    out[3] = l_span;
    out[4] = l_cons;
  }
}

// ---------------------------------------------------------------------------
extern "C" void kernel_launch(void* const* d_in, const int* in_sizes, int n_in,
                              void* d_out, int out_size, void* d_ws,
                              size_t ws_size, hipStream_t stream) {
  (void)in_sizes; (void)n_in; (void)out_size; (void)ws_size;
  const float* en    = (const float*)d_in[0];
  const float* vi    = (const float*)d_in[1];
  const float* gamma = (const float*)d_in[2];
  const float* Den   = (const float*)d_in[3];
  const float* Dvi   = (const float*)d_in[4];
  const float* M     = (const float*)d_in[5];
  const int*   es    = (const int*)d_in[6];
  const int*   ee    = (const int*)d_in[7];
  const float* w_s   = (const float*)d_in[8];
  const float* b_s   = (const float*)d_in[9];
  const float* w_e   = (const float*)d_in[10];
  const float* b_e   = (const float*)d_in[11];
  float* ws  = (float*)d_ws;
  float* out = (float*)d_out;

  qa_logits_kernel<<<(2 * BK) / 8, 256, 0, stream>>>(en, vi, w_s, b_s, w_e,
                                                     b_e, ws);
  fgw_kernel<<<BB, 256, 0, stream>>>(gamma, Den, Dvi, M, ws);
  span_kernel<<<BB, 256, 0, stream>>>(gamma, es, ee, ws);
  finalize_kernel<<<1, 256, 0, stream>>>(ws, out);
}